// GraphConvolution_6502580486593
// MI455X (gfx1250) — compile-verified
//
#include <hip/hip_runtime.h>

typedef __attribute__((ext_vector_type(2))) float v2f;
typedef __attribute__((ext_vector_type(8))) float v8f;
typedef __attribute__((ext_vector_type(4))) float v4f;

#define IN_F 128
#define OUT_F 64
#define WPB 4                    // waves per block
#define XPAD 4                   // pad row stride to 132 floats (16B-aligned, bank-conflict-free)
#define XSTRIDE (IN_F + XPAD)

// ---------------------------------------------------------------------------
// Kernel 1: support = x @ W via V_WMMA_F32_16X16X4_F32 (exact fp32 path).
// One wave computes a 16(M) x 64(N) strip. The 16x128 x-tile is staged through
// a wave-private LDS slab with fully-coalesced b128 global loads; LDS ops are
// in-order per wave so no barrier is required.
// ---------------------------------------------------------------------------
__global__ __launch_bounds__(WPB * 32) void gcn_gemm_wmma(
    const float* __restrict__ x, const float* __restrict__ w,
    float* __restrict__ support, int n_nodes)
{
    __shared__ float sX[WPB][16 * XSTRIDE];   // 4 * 2112 floats = 33 KB

    const int lane = threadIdx.x & 31;
    const int wid  = threadIdx.x >> 5;
    const int wave = blockIdx.x * WPB + wid;
    const int row0 = wave << 4;               // 16 node-rows per wave
    if (row0 >= n_nodes) return;              // wave-uniform: EXEC stays all-1s

    // ---- stage x tile: one full 512B row per 32-lane b128 batch ----
    float* __restrict__ slab = &sX[wid][0];
    #pragma unroll 4
    for (int r = 0; r < 16; ++r) {
        v4f t = *(const v4f*)(x + (size_t)(row0 + r) * IN_F + (lane << 2));
        *(v4f*)(slab + r * XSTRIDE + (lane << 2)) = t;
    }

    const int m     = lane & 15;              // A: lane -> M row
    const int n     = lane & 15;              // B/C/D: lane -> N col
    const int khalf = (lane >> 4) << 1;       // lanes 16-31 hold K+2,K+3

    const float* __restrict__ axrow = slab + m * XSTRIDE + khalf;

    v8f acc0 = {}; v8f acc1 = {}; v8f acc2 = {}; v8f acc3 = {};

    #pragma unroll 4
    for (int k = 0; k < IN_F; k += 4) {
        // A fragment from LDS: contiguous 8B per lane, conflict-free banks
        v2f a;
        a.x = axrow[k];
        a.y = axrow[k + 1];

        // B fragments: rows k+khalf, k+khalf+1 of W (L0-resident 32KB)
        const float* __restrict__ w0 = w + (size_t)(k + khalf) * OUT_F + n;
        const float* __restrict__ w1 = w0 + OUT_F;
        v2f b0; b0.x = w0[0];  b0.y = w1[0];
        v2f b1; b1.x = w0[16]; b1.y = w1[16];
        v2f b2; b2.x = w0[32]; b2.y = w1[32];
        v2f b3; b3.x = w0[48]; b3.y = w1[48];

        acc0 = __builtin_amdgcn_wmma_f32_16x16x4_f32(false, a, false, b0, (short)0, acc0, false, false);
        acc1 = __builtin_amdgcn_wmma_f32_16x16x4_f32(false, a, false, b1, (short)0, acc1, false, false);
        acc2 = __builtin_amdgcn_wmma_f32_16x16x4_f32(false, a, false, b2, (short)0, acc2, false, false);
        acc3 = __builtin_amdgcn_wmma_f32_16x16x4_f32(false, a, false, b3, (short)0, acc3, false, false);
    }

    // C/D layout: VGPR i -> M = i + 8*(lane>=16), N = lane%16
    const int mout = (lane >> 4) << 3;
    float* __restrict__ outb = support + (size_t)row0 * OUT_F;
    #pragma unroll
    for (int i = 0; i < 8; ++i) {
        float* r = outb + (size_t)(mout + i) * OUT_F + n;
        r[0]  = acc0[i];
        r[16] = acc1[i];
        r[32] = acc2[i];
        r[48] = acc3[i];
    }
}

// ---------------------------------------------------------------------------
// Kernel 0: zero the output accumulator (d_out is poisoned by the harness).
// ---------------------------------------------------------------------------
__global__ __launch_bounds__(256) void gcn_zero(float* __restrict__ out, int n4)
{
    int i = blockIdx.x * blockDim.x + threadIdx.x;
    v4f z = {};
    if (i < n4) ((v4f*)out)[i] = z;
}

// ---------------------------------------------------------------------------
// Kernel 2: COO SpMM scatter: out[row] += val * support[col].
// One wave per edge; 32 lanes cover 64 features (2 atomics each).
// support & out both live in the 192MB L2, so atomics stay on-chip.
// ---------------------------------------------------------------------------
__global__ __launch_bounds__(256) void gcn_spmm(
    const float* __restrict__ support,
    const int*   __restrict__ erow, const int* __restrict__ ecol,
    const float* __restrict__ eval, float* __restrict__ out, int n_edges)
{
    const int lane = threadIdx.x & 31;
    const int e    = blockIdx.x * (blockDim.x >> 5) + (threadIdx.x >> 5);
    if (e >= n_edges) return;

    // prefetch the next batch of the streamed edge arrays (speculative, safe)
    __builtin_prefetch(&ecol[e + 2048], 0, 1);
    __builtin_prefetch(&erow[e + 2048], 0, 1);

    const int   r = erow[e];       // wave-uniform broadcast loads
    const int   c = ecol[e];
    const float v = eval[e];

    const float* __restrict__ s = support + (size_t)c * OUT_F;
    float*       __restrict__ o = out     + (size_t)r * OUT_F;

    atomicAdd(&o[lane],      v * s[lane]);
    atomicAdd(&o[lane + 32], v * s[lane + 32]);
}

// ---------------------------------------------------------------------------
// Kernel 3: in-place ReLU.
// ---------------------------------------------------------------------------
__global__ __launch_bounds__(256) void gcn_relu(float* __restrict__ out, int n4)
{
    int i = blockIdx.x * blockDim.x + threadIdx.x;
    if (i < n4) {
        v4f t = ((v4f*)out)[i];
        t.x = fmaxf(t.x, 0.f); t.y = fmaxf(t.y, 0.f);
        t.z = fmaxf(t.z, 0.f); t.w = fmaxf(t.w, 0.f);
        ((v4f*)out)[i] = t;
    }
}

extern "C" void kernel_launch(void* const* d_in, const int* in_sizes, int n_in,
                              void* d_out, int out_size, void* d_ws, size_t ws_size,
                              hipStream_t stream)
{
    const float* x      = (const float*)d_in[0];
    const float* weight = (const float*)d_in[1];
    const int*   erow   = (const int*)  d_in[2];
    const int*   ecol   = (const int*)  d_in[3];
    const float* eval   = (const float*)d_in[4];
    float*       out    = (float*)d_out;

    const int n_nodes = in_sizes[0] / IN_F;     // 100000
    const int n_edges = in_sizes[2];            // 1600000

    float* support = (float*)d_ws;              // n_nodes * 64 * 4 = 25.6 MB scratch

    // 1) zero accumulator
    {
        int n4 = out_size / 4;
        gcn_zero<<<(n4 + 255) / 256, 256, 0, stream>>>(out, n4);
    }
    // 2) dense GEMM via WMMA f32 (4 waves/block, 16 rows/wave, LDS-staged A)
    {
        int waves  = n_nodes / 16;              // 6250 (N_NODES % 16 == 0)
        int blocks = (waves + WPB - 1) / WPB;
        gcn_gemm_wmma<<<blocks, WPB * 32, 0, stream>>>(x, weight, support, n_nodes);
    }
    // 3) edge scatter (1 wave per edge, 8 edges/block)
    {
        int blocks = (n_edges + 7) / 8;
        gcn_spmm<<<blocks, 256, 0, stream>>>(support, erow, ecol, eval, out, n_edges);
    }
    // 4) ReLU in place
    {
        int n4 = out_size / 4;
        gcn_relu<<<(n4 + 255) / 256, 256, 0, stream>>>(out, n4);
    }
}